// SelfLinAttention_5660766896668
// MI455X (gfx1250) — compile-verified
//
#include <hip/hip_runtime.h>
#include <hip/hip_bf16.h>

// ---------------------------------------------------------------------------
// Causal linear attention (elu+1 feature map) for (B=2, S=2048, H=16, D=64),
// chunked-scan formulation, f32 WMMA (v_wmma_f32_16x16x4_f32), wave32.
// ---------------------------------------------------------------------------

typedef float v2f __attribute__((ext_vector_type(2)));
typedef float v8f __attribute__((ext_vector_type(8)));

constexpr int B_   = 2;
constexpr int S_   = 2048;
constexpr int H_   = 16;
constexpr int D_   = 64;
constexpr int CH   = 64;      // chunk (tokens)
constexpr int LDP  = 68;      // padded LDS row stride in floats (4-float pad)
constexpr int NSPL = 8;       // sequence splits (parallel prefix over KV state)

// dynamic LDS layout (floats)
constexpr int OFF_Q    = 0;
constexpr int OFF_K    = 64 * LDP;        // 4352
constexpr int OFF_V    = 2 * 64 * LDP;    // 8704
constexpr int OFF_S    = 3 * 64 * LDP;    // 13056
constexpr int OFF_KV   = 4 * 64 * LDP;    // 17408
constexpr int OFF_KSUM = 5 * 64 * LDP;    // 21760
constexpr int OFF_DEN  = OFF_KSUM + 64;   // 21824
constexpr int SMEM_F   = OFF_DEN + 64;    // 21888 floats = 87552 B (<320KB/WG)

constexpr int STATE_F  = 64 * 64 + 64;    // per-(b,h,split) ws state: KV + ksum

__device__ __forceinline__ float elu1(float x) {
  // elu(x)+1 : x>0 ? x+1 : exp(x)
  return x > 0.0f ? x + 1.0f : __expf(x);
}

__device__ __forceinline__ v8f wmma4(v2f a, v2f b, v8f c) {
  // D = A(16x4,f32) * B(4x16,f32) + C(16x16,f32)
  return __builtin_amdgcn_wmma_f32_16x16x4_f32(
      /*neg_a=*/false, a, /*neg_b=*/false, b,
      /*c_mod=*/(short)0, c, /*reuse_a=*/false, /*reuse_b=*/false);
}

// 8-byte aligned LDS pair load (emits ds_load_b64): idx must be even.
__device__ __forceinline__ v2f lds_pair(const float* p) {
  return *reinterpret_cast<const v2f*>(p);
}

// ---- C/D-layout (ISA 7.12.2) LDS helpers: VGPR r -> M = r + 8*(lane>=16) ---
__device__ __forceinline__ v8f cd_load(const float* base, int mi, int nj, int lane) {
  int n  = nj * 16 + (lane & 15);
  int m0 = mi * 16 + ((lane >> 4) << 3);
  v8f c;
#pragma unroll
  for (int r = 0; r < 8; ++r) c[r] = base[(m0 + r) * LDP + n];
  return c;
}
__device__ __forceinline__ void cd_store(float* base, int mi, int nj, int lane, v8f c) {
  int n  = nj * 16 + (lane & 15);
  int m0 = mi * 16 + ((lane >> 4) << 3);
#pragma unroll
  for (int r = 0; r < 8; ++r) base[(m0 + r) * LDP + n] = c[r];
}

// ---- GEMM tile helpers: acc(16x16) over k-range [k0,k1) of 4-wide steps ----
// A row-major (M x K), B row-major (K x N):   acc += A * B
__device__ __forceinline__ v8f gemm_nn(const float* A, const float* Bm,
                                       int mi, int nj, int lane,
                                       int k0, int k1, v8f acc) {
  int lm = lane & 15, hi = lane >> 4;
  int arow = (mi * 16 + lm) * LDP;
  int bcol = nj * 16 + lm;
  for (int kk = k0; kk < k1; ++kk) {
    int kb = kk * 4 + hi * 2;
    v2f a = lds_pair(&A[arow + kb]);                           // ds_load_b64
    v2f b; b.x = Bm[kb * LDP + bcol]; b.y = Bm[(kb + 1) * LDP + bcol];
    acc = wmma4(a, b, acc);
  }
  return acc;
}
// A row-major (M x K), B accessed transposed (N x K storage): acc += A * B^T
__device__ __forceinline__ v8f gemm_nt(const float* A, const float* Bt,
                                       int mi, int nj, int lane, v8f acc) {
  int lm = lane & 15, hi = lane >> 4;
  int arow = (mi * 16 + lm) * LDP;
  int brow = (nj * 16 + lm) * LDP;
#pragma unroll
  for (int kk = 0; kk < 16; ++kk) {
    int kb = kk * 4 + hi * 2;
    v2f a = lds_pair(&A[arow + kb]);                           // ds_load_b64
    v2f b = lds_pair(&Bt[brow + kb]);                          // ds_load_b64
    acc = wmma4(a, b, acc);
  }
  return acc;
}
// A accessed transposed (K x M storage), B row-major: acc += A^T * B
__device__ __forceinline__ v8f gemm_tn(const float* At, const float* Bm,
                                       int mi, int nj, int lane, v8f acc) {
  int lm = lane & 15, hi = lane >> 4;
  int am   = mi * 16 + lm;
  int bcol = nj * 16 + lm;
#pragma unroll
  for (int kk = 0; kk < 16; ++kk) {
    int kb = kk * 4 + hi * 2;
    v2f a; a.x = At[kb * LDP + am];   a.y = At[(kb + 1) * LDP + am];
    v2f b; b.x = Bm[kb * LDP + bcol]; b.y = Bm[(kb + 1) * LDP + bcol];
    acc = wmma4(a, b, acc);
  }
  return acc;
}

__device__ __forceinline__ void mask_diag(v8f& c, int lane) {
  int n  = lane & 15;
  int m0 = (lane >> 4) << 3;
#pragma unroll
  for (int r = 0; r < 8; ++r)
    if (m0 + r < n) c[r] = 0.0f;
}

// load 64x64 f32 tile (global, rowStride floats) -> padded LDS, optional elu+1
__device__ __forceinline__ void load_tile(const float* __restrict__ g, int rowStride,
                                          float* lds, int tid, bool do_elu) {
#pragma unroll
  for (int j = 0; j < 4; ++j) {
    int u = tid + j * 256;
    int r = u >> 4;
    int c = (u & 15) << 2;
    float4 w = *reinterpret_cast<const float4*>(g + (size_t)r * rowStride + c);
    if (do_elu) { w.x = elu1(w.x); w.y = elu1(w.y); w.z = elu1(w.z); w.w = elu1(w.w); }
    *reinterpret_cast<float4*>(lds + r * LDP + c) = w;
  }
}

// ---------------------------------------------------------------------------
// Pass 1: per-(b,h,split) local state  KV = sum K^T V,  ksum = sum K rows
// ---------------------------------------------------------------------------
__global__ __launch_bounds__(256) void lin_attn_states(
    const float* __restrict__ qk, const float* __restrict__ v,
    float* __restrict__ ws, int nsplit) {
  extern __shared__ float smem[];
  float* sK    = smem + OFF_K;
  float* sV    = smem + OFF_V;
  float* sKV   = smem + OFF_KV;
  float* sksum = smem + OFF_KSUM;

  int tid = threadIdx.x, lane = tid & 31, wid = tid >> 5;
  int wg = blockIdx.x;
  int split = wg % nsplit;
  int h     = (wg / nsplit) % H_;
  int b     = wg / (nsplit * H_);
  int cps   = (S_ / CH) / nsplit;

  for (int i = tid; i < 64 * LDP; i += 256) sKV[i] = 0.0f;
  if (tid < 64) sksum[tid] = 0.0f;
  __syncthreads();

  for (int c = 0; c < cps; ++c) {
    int base = (split * cps + c) * CH;
    const float* kptr = qk + (((size_t)(b * S_ + base) * 2 + 1) * H_ + h) * D_;
    const float* vptr = v  + (((size_t)(b * S_ + base)) * H_ + h) * D_;
    load_tile(kptr, 2 * H_ * D_, sK, tid, true);
    load_tile(vptr, H_ * D_,     sV, tid, false);
    __syncthreads();

    if (tid < 64) {                      // ksum += column sums of K  (waves 0,1)
      float acc = 0.0f;
      for (int s = 0; s < 64; ++s) acc += sK[s * LDP + tid];
      sksum[tid] += acc;
    }
#pragma unroll
    for (int i = 0; i < 2; ++i) {        // KV += K^T V  (16 tiles, 2 per wave)
      int t = wid * 2 + i, mi = t >> 2, nj = t & 3;
      v8f acc = cd_load(sKV, mi, nj, lane);
      acc = gemm_tn(sK, sV, mi, nj, lane, acc);
      cd_store(sKV, mi, nj, lane, acc);
    }
    __syncthreads();
  }

  float* wsp = ws + (size_t)((b * H_ + h) * nsplit + split) * STATE_F;
  for (int i = tid; i < 64 * 64; i += 256)
    wsp[i] = sKV[(i >> 6) * LDP + (i & 63)];
  if (tid < 64) wsp[64 * 64 + tid] = sksum[tid];
}

// ---------------------------------------------------------------------------
// Pass 2: prefix states + chunk scan + masked intra-chunk attention + output
// ---------------------------------------------------------------------------
__global__ __launch_bounds__(256) void lin_attn_main(
    const float* __restrict__ qk, const float* __restrict__ v,
    const float* __restrict__ ws, float* __restrict__ out, int nsplit) {
  extern __shared__ float smem[];
  float* sQ    = smem + OFF_Q;
  float* sK    = smem + OFF_K;
  float* sV    = smem + OFF_V;
  float* sS    = smem + OFF_S;
  float* sKV   = smem + OFF_KV;
  float* sksum = smem + OFF_KSUM;
  float* sden  = smem + OFF_DEN;

  int tid = threadIdx.x, lane = tid & 31, wid = tid >> 5;
  int wg = blockIdx.x;
  int split = wg % nsplit;
  int h     = (wg / nsplit) % H_;
  int b     = wg / (nsplit * H_);
  int cps   = (S_ / CH) / nsplit;

  for (int i = tid; i < 64 * LDP; i += 256) sKV[i] = 0.0f;
  if (tid < 64) sksum[tid] = 0.0f;
  __syncthreads();

  // exclusive prefix of predecessor splits' states
  for (int p = 0; p < split; ++p) {
    const float* wsp = ws + (size_t)((b * H_ + h) * nsplit + p) * STATE_F;
    for (int i = tid; i < 64 * 64; i += 256)
      sKV[(i >> 6) * LDP + (i & 63)] += wsp[i];
    if (tid < 64) sksum[tid] += wsp[64 * 64 + tid];
  }
  __syncthreads();

  int t0 = wid * 2, t1 = wid * 2 + 1;
  int m0 = t0 >> 2, n0 = t0 & 3, m1 = t1 >> 2, n1 = t1 & 3;

  for (int c = 0; c < cps; ++c) {
    int base = (split * cps + c) * CH;
    const float* qptr = qk + (((size_t)(b * S_ + base) * 2 + 0) * H_ + h) * D_;
    const float* kptr = qk + (((size_t)(b * S_ + base) * 2 + 1) * H_ + h) * D_;
    const float* vptr = v  + (((size_t)(b * S_ + base)) * H_ + h) * D_;
    load_tile(qptr, 2 * H_ * D_, sQ, tid, true);
    load_tile(kptr, 2 * H_ * D_, sK, tid, true);
    load_tile(vptr, H_ * D_,     sV, tid, false);

    if (c + 1 < cps) {                   // global_prefetch_b8 next chunk
      int nb = base + CH;
      const float* nq = qk + (((size_t)(b * S_ + nb) * 2) * H_ + h) * D_;
      const float* nv = v  + (((size_t)(b * S_ + nb)) * H_ + h) * D_;
      int r = tid >> 2, o = (tid & 3) << 4;
      __builtin_prefetch(nq + (size_t)r * 2 * H_ * D_ + o, 0, 3);
      __builtin_prefetch(nq + (size_t)r * 2 * H_ * D_ + H_ * D_ + o, 0, 3);
      __builtin_prefetch(nv + (size_t)r * H_ * D_ + o, 0, 3);
    }
    __syncthreads();

    // (1) inter-chunk numerator: OUT = Q @ KV_prev
    v8f oa0 = {}, oa1 = {};
    oa0 = gemm_nn(sQ, sKV, m0, n0, lane, 0, 16, oa0);
    oa1 = gemm_nn(sQ, sKV, m1, n1, lane, 0, 16, oa1);

    // (1b) inter-chunk denominator: q_t . ksum_prev   (waves 0,1)
    if (tid < 64) {
      float d = 0.0f;
      for (int dd = 0; dd < 64; ++dd) d += sQ[tid * LDP + dd] * sksum[dd];
      sden[tid] = d;
    }

    // (2) intra-chunk masked scores S = tril(Q @ K^T) -> LDS
    if (n0 <= m0) {
      v8f sa = {};
      sa = gemm_nt(sQ, sK, m0, n0, lane, sa);
      if (m0 == n0) mask_diag(sa, lane);
      cd_store(sS, m0, n0, lane, sa);
    }
    if (n1 <= m1) {
      v8f sa = {};
      sa = gemm_nt(sQ, sK, m1, n1, lane, sa);
      if (m1 == n1) mask_diag(sa, lane);
      cd_store(sS, m1, n1, lane, sa);
    }
    __syncthreads();

    // (2b) denom = 1 / (inter + row-sum of masked scores)
    if (tid < 64) {
      float ssum = 0.0f;
      int smax = ((tid >> 4) + 1) * 16;  // masked zeros cover s>t inside diag tile
      for (int s = 0; s < smax; ++s) ssum += sS[tid * LDP + s];
      sden[tid] = 1.0f / (sden[tid] + ssum);
    }
    __syncthreads();

    // (3) OUT += S @ V  (triangular k-range per row-block)
    oa0 = gemm_nn(sS, sV, m0, n0, lane, 0, (m0 + 1) * 4, oa0);
    oa1 = gemm_nn(sS, sV, m1, n1, lane, 0, (m1 + 1) * 4, oa1);

    // (3b) scale by 1/denom and store
    {
      int nn = n0 * 16 + (lane & 15);
      int mm = m0 * 16 + ((lane >> 4) << 3);
#pragma unroll
      for (int r = 0; r < 8; ++r)
        out[(((size_t)b * S_ + base + mm + r) * H_ + h) * D_ + nn] = oa0[r] * sden[mm + r];
      nn = n1 * 16 + (lane & 15);
      mm = m1 * 16 + ((lane >> 4) << 3);
#pragma unroll
      for (int r = 0; r < 8; ++r)
        out[(((size_t)b * S_ + base + mm + r) * H_ + h) * D_ + nn] = oa1[r] * sden[mm + r];
    }
    __syncthreads();

    // (4) state update: KV += K^T V ; ksum += col-sums of K
    if (tid < 64) {
      float acc = 0.0f;
      for (int s = 0; s < 64; ++s) acc += sK[s * LDP + tid];
      sksum[tid] += acc;
    }
    {
      v8f a0 = cd_load(sKV, m0, n0, lane);
      a0 = gemm_tn(sK, sV, m0, n0, lane, a0);
      cd_store(sKV, m0, n0, lane, a0);
      v8f a1 = cd_load(sKV, m1, n1, lane);
      a1 = gemm_tn(sK, sV, m1, n1, lane, a1);
      cd_store(sKV, m1, n1, lane, a1);
    }
    __syncthreads();
  }
}

// ---------------------------------------------------------------------------
extern "C" void kernel_launch(void* const* d_in, const int* in_sizes, int n_in,
                              void* d_out, int out_size, void* d_ws, size_t ws_size,
                              hipStream_t stream) {
  const float* qk = (const float*)d_in[0];
  const float* v  = (const float*)d_in[1];
  float* out = (float*)d_out;
  float* ws  = (float*)d_ws;

  int nsplit = NSPL;
  size_t need = (size_t)B_ * H_ * NSPL * STATE_F * sizeof(float);
  if (ws_size < need) nsplit = 1;     // fallback: single pass, no workspace

  int wgs = B_ * H_ * nsplit;
  size_t smem = (size_t)SMEM_F * sizeof(float);

  if (nsplit > 1)
    lin_attn_states<<<wgs, 256, smem, stream>>>(qk, v, ws, nsplit);
  lin_attn_main<<<wgs, 256, smem, stream>>>(qk, v, ws, out, nsplit);
}